// ROIPooler_81810537055085
// MI455X (gfx1250) — compile-verified
//
#include <hip/hip_runtime.h>

// ROIAlign FPN pooler for gfx1250 (MI455X), using the CDNA5 Tensor Data Mover
// (TENSOR_LOAD_TO_LDS with descriptor iteration) to stage per-box feature
// regions for 16 channels into LDS with one async DMA instruction, then doing
// the bilinear gather from LDS. TDM issue is overlapped with the per-pixel
// sample-geometry computation; output stores are non-temporal so the write-
// once result does not evict the L2-resident feature pyramid.

#define ROI_OUT   14
#define CTILE     16
#define RPITCH    1216        // floats per channel slot in LDS (max region ~1130)
#define C_TOTAL   256
#define BASE_HW   256

typedef __attribute__((ext_vector_type(4))) unsigned int u32x4;
typedef __attribute__((ext_vector_type(8))) unsigned int u32x8;

__device__ __forceinline__ unsigned rfl(unsigned x) {
  return (unsigned)__builtin_amdgcn_readfirstlane((int)x);
}

__global__ __launch_bounds__(224) void roi_align_tdm_kernel(
    const float* __restrict__ f0, const float* __restrict__ f1,
    const float* __restrict__ f2, const float* __restrict__ f3,
    const float* __restrict__ boxes, const int* __restrict__ bidx,
    float* __restrict__ out)
{
  __shared__ float smem[CTILE * RPITCH];

  const int box = blockIdx.y;
  const int c0  = blockIdx.x * CTILE;
  const int t   = threadIdx.x;

  // ---- uniform per-block geometry (scalarized by compiler) ------------
  const float bx0 = boxes[box * 4 + 0];
  const float by0 = boxes[box * 4 + 1];
  const float bx1 = boxes[box * 4 + 2];
  const float by1 = boxes[box * 4 + 3];
  const int   n   = bidx[box];

  // FPN level assignment (matches reference _assign_levels)
  const float area = fmaxf((bx1 - bx0) * (by1 - by0), 0.0f);
  const float size = sqrtf(area);
  float lv = floorf(4.0f + log2f(size * (1.0f / 224.0f) + 2.220446049250313e-16f));
  lv = fminf(fmaxf(lv, 2.0f), 5.0f);
  const int l = (int)lv - 2;                 // 0..3

  const float* feat = (l == 0) ? f0 : (l == 1) ? f1 : (l == 2) ? f2 : f3;
  const int W = BASE_HW >> l;
  const int H = W;
  const float scale = 0.25f / (float)(1 << l);
  const float fW1 = (float)(W - 1);
  const float fH1 = (float)(H - 1);

  const float x0f = bx0 * scale - 0.5f;
  const float x1f = bx1 * scale - 0.5f;
  const float y0f = by0 * scale - 0.5f;
  const float y1f = by1 * scale - 0.5f;
  const float bw = (x1f - x0f) * (1.0f / (float)ROI_OUT);
  const float bh = (y1f - y0f) * (1.0f / (float)ROI_OUT);

  // sample coordinate extremes: g in {0.25 ... 13.75}, bw/bh > 0
  const float sxmin = x0f + 0.25f * bw, sxmax = x0f + 13.75f * bw;
  const float symin = y0f + 0.25f * bh, symax = y0f + 13.75f * bh;

  const int xr0 = (int)floorf(fminf(fmaxf(sxmin, 0.0f), fW1));
  int       xr1 = (int)floorf(fminf(fmaxf(sxmax, 0.0f), fW1)) + 1;
  const int yr0 = (int)floorf(fminf(fmaxf(symin, 0.0f), fH1));
  int       yr1 = (int)floorf(fminf(fmaxf(symax, 0.0f), fH1)) + 1;
  if (xr1 > W - 1) xr1 = W - 1;
  if (yr1 > H - 1) yr1 = H - 1;
  const int Wr = xr1 - xr0 + 1;
  int       Hr = yr1 - yr0 + 1;
  if (Wr * Hr > RPITCH) Hr = RPITCH / Wr;    // defensive; unreachable by level math

  const bool issuer = (t < 32);              // wave 0 owns the TDM

  // ---- TDM issue: stage CTILE channel regions into LDS ----------------
  if (issuer) {
    const unsigned HW = (unsigned)(H * W);
    const unsigned long long elem =
        ((unsigned long long)(unsigned)(n * C_TOTAL + c0) * (unsigned)H + (unsigned)yr0)
            * (unsigned)W + (unsigned)xr0;
    const unsigned long long ga = (unsigned long long)(const void*)feat + 4ull * elem;
    const unsigned ldsb = (unsigned)(unsigned long long)(const void*)&smem[0];

    u32x4 g0; u32x8 g1; u32x4 g2; u32x4 g3;
    // Group 0: count=1 | lds_addr | global_addr[56:0] | type=2 (bits 127:126)
    g0[0] = rfl(1u);
    g0[1] = rfl(ldsb);
    g0[2] = rfl((unsigned)(ga & 0xFFFFFFFFull));
    g0[3] = rfl(((unsigned)((ga >> 32) & 0x01FFFFFFull)) | 0x80000000u);
    // Group 1: data_size=2 (4B), iterate_enable=1; tensor dims/strides; tile dims
    g1[0] = rfl((2u << 16) | (1u << 19));
    g1[1] = rfl(((unsigned)W & 0xFFFFu) << 16);                 // tensor_dim0[15:0]
    g1[2] = rfl((((unsigned)W >> 16) & 0xFFFFu) |
                (((unsigned)H & 0xFFFFu) << 16));               // t_dim0[31:16] | t_dim1[15:0]
    g1[3] = rfl((((unsigned)H >> 16) & 0xFFFFu) |
                (((unsigned)Wr & 0xFFFFu) << 16));              // t_dim1[31:16] | tile_dim0
    g1[4] = rfl((unsigned)Hr & 0xFFFFu);                        // tile_dim1, tile_dim2=0
    g1[5] = rfl((unsigned)W);                                   // tensor_dim0_stride lo32
    g1[6] = rfl((HW & 0xFFFFu) << 16);                          // t0s hi16=0 | t1s[15:0]
    g1[7] = rfl(HW >> 16);                                      // t1s[47:16]
    // Group 2 (iterate mode): tensor_dim2 | lds_incr | global_incr | iter_count
    g2[0] = rfl(0x40000000u);                                   // tensor_dim2 (large)
    g2[1] = rfl((unsigned)RPITCH);                              // lds_addr_increment (elems)
    g2[2] = rfl(HW);                                            // global_addr_increment lo32
    g2[3] = rfl(((unsigned)(CTILE - 1)) << 16);                 // gai hi16=0 | iterate_count
    g3[0] = rfl(0u); g3[1] = rfl(0u); g3[2] = rfl(0u); g3[3] = rfl(0u);

    asm volatile("tensor_load_to_lds %0, %1, %2, %3"
                 :: "s"(g0), "s"(g1), "s"(g2), "s"(g3)
                 : "memory");
    // NOTE: no wait here — overlap the DMA with sample-geometry setup below.
  }

  // ---- per-pixel sample geometry (independent of LDS contents) --------
  const bool active = (t < ROI_OUT * ROI_OUT);
  const int by = t / ROI_OUT;
  const int bx = t - by * ROI_OUT;

  int   off[4], dx[4], dyW[4];
  float w00[4], w01[4], w10[4], w11[4];
  if (active) {
#pragma unroll
    for (int k = 0; k < 4; ++k) {
      const int i = k >> 1;        // y sub-sample
      const int j = k & 1;         // x sub-sample
      const float ysmp = y0f + ((float)by + ((float)i + 0.5f) * 0.5f) * bh;
      const float xsmp = x0f + ((float)bx + ((float)j + 0.5f) * 0.5f) * bw;
      const bool valid = (ysmp > -1.0f) && (ysmp < (float)H) &&
                         (xsmp > -1.0f) && (xsmp < (float)W);
      const float yc = fminf(fmaxf(ysmp, 0.0f), fH1);
      const float xc = fminf(fmaxf(xsmp, 0.0f), fW1);
      const int yl = (int)floorf(yc);
      const int xl = (int)floorf(xc);
      const int yh = min(yl + 1, H - 1);
      const int xh = min(xl + 1, W - 1);
      const float ly = yc - (float)yl;
      const float lx = xc - (float)xl;
      const float hy = 1.0f - ly;
      const float hx = 1.0f - lx;
      const float m = valid ? 1.0f : 0.0f;
      w00[k] = hy * hx * m;
      w01[k] = hy * lx * m;
      w10[k] = ly * hx * m;
      w11[k] = ly * lx * m;
      off[k] = (yl - yr0) * Wr + (xl - xr0);
      dx[k]  = xh - xl;
      dyW[k] = (yh - yl) * Wr;
    }
  }

  // ---- drain the DMA, then publish LDS to all waves -------------------
  if (issuer) {
    __builtin_amdgcn_s_wait_tensorcnt(0);
  }
  __syncthreads();

  // ---- bilinear gather from LDS: one thread per output pixel ----------
  if (active) {
    const long long obase =
        ((long long)box * C_TOTAL + c0) * (ROI_OUT * ROI_OUT) +
        (long long)(by * ROI_OUT + bx);

#pragma unroll 4
    for (int c = 0; c < CTILE; ++c) {
      const float* L = &smem[c * RPITCH];
      float acc = 0.0f;
#pragma unroll
      for (int k = 0; k < 4; ++k) {
        const float v00 = L[off[k]];
        const float v01 = L[off[k] + dx[k]];
        const float v10 = L[off[k] + dyW[k]];
        const float v11 = L[off[k] + dyW[k] + dx[k]];
        acc += w00[k] * v00 + w01[k] * v01 + w10[k] * v10 + w11[k] * v11;
      }
      // write-once result: non-temporal so it doesn't evict L2-resident features
      __builtin_nontemporal_store(0.25f * acc,
                                  &out[obase + (long long)c * (ROI_OUT * ROI_OUT)]);
    }
  }
}

extern "C" void kernel_launch(void* const* d_in, const int* in_sizes, int n_in,
                              void* d_out, int out_size, void* d_ws, size_t ws_size,
                              hipStream_t stream) {
  (void)n_in; (void)out_size; (void)d_ws; (void)ws_size;
  const float* f0    = (const float*)d_in[0];
  const float* f1    = (const float*)d_in[1];
  const float* f2    = (const float*)d_in[2];
  const float* f3    = (const float*)d_in[3];
  const float* boxes = (const float*)d_in[4];
  const int*   bidx  = (const int*)d_in[5];
  float*       out   = (float*)d_out;

  const int M = in_sizes[4] / 4;                 // 256 boxes
  dim3 grid(C_TOTAL / CTILE, M);                 // (16, 256)
  roi_align_tdm_kernel<<<grid, 224, 0, stream>>>(f0, f1, f2, f3, boxes, bidx, out);
}